// NodeMaxAggregator_73469710565690
// MI455X (gfx1250) — compile-verified
//
#include <hip/hip_runtime.h>

// NodeMaxAggregator: per-node gather of 16 hyperedge embeddings (dim 128, fp32)
// from a 200k-row table, elementwise max-pooled per node.
//
// MI455X reasoning:
//  - Table (102.4 MB) fits in 192 MB L2 -> keep table loads default (RT) so the
//    819 MB of gather traffic stays L2-resident.
//  - Output (51 MB) is write-once -> non-temporal store (TH=NT) so it does not
//    evict the table from L2.
//  - wave32: one wave per node; 128 floats = 32 lanes x float4 -> each row read
//    is a single coalesced global_load_b128 per wave.
//  - All 16 row loads issued back-to-back (fully unrolled, indices preloaded)
//    for maximum memory-level parallelism.

typedef float v4f __attribute__((ext_vector_type(4)));

#define EMBED_DIM       128
#define DEG             16
#define WAVES_PER_BLOCK 8
#define BLOCK_THREADS   (WAVES_PER_BLOCK * 32)

__global__ __launch_bounds__(BLOCK_THREADS) void node_max_aggregate_kernel(
    const float* __restrict__ table,   // [N_HYPEREDGES, 128]
    const int*   __restrict__ ids,     // [n_nodes * DEG] hyperedge ids (int32)
    float*       __restrict__ out,     // [n_nodes, 128]
    int n_nodes)
{
    // Wave id is uniform across the wave; readfirstlane makes that provable so
    // the node-index math (and the index-block loads below) become scalar.
    const int wave = __builtin_amdgcn_readfirstlane((int)(threadIdx.x >> 5));
    const int lane = (int)(threadIdx.x & 31u);
    const int node = (int)blockIdx.x * WAVES_PER_BLOCK + wave;
    if (node >= n_nodes) return;   // wave-uniform exit

    // Preload this node's 16 hyperedge ids (wave-uniform address -> s_load
    // candidates). Keeping them in registers lets all 16 row loads issue
    // without interleaved dependent index fetches.
    const int* __restrict__ nid = ids + (size_t)node * DEG;
    int e[DEG];
#pragma unroll
    for (int k = 0; k < DEG; ++k) {
        e[k] = nid[k];
    }

    // Max-accumulate 16 rows. Each lane owns columns [4*lane, 4*lane+4):
    // one global_load_b128 per lane per row; 512B per wave per row, coalesced.
    const float NEG_INF = -__builtin_huge_valf();
    v4f acc = (v4f){NEG_INF, NEG_INF, NEG_INF, NEG_INF};

#pragma unroll
    for (int k = 0; k < DEG; ++k) {
        const v4f* __restrict__ row =
            (const v4f*)(table + (size_t)e[k] * EMBED_DIM);
        v4f v = row[lane];                 // global_load_b128 (RT -> L2 resident)
        acc.x = __builtin_fmaxf(acc.x, v.x);
        acc.y = __builtin_fmaxf(acc.y, v.y);
        acc.z = __builtin_fmaxf(acc.z, v.z);
        acc.w = __builtin_fmaxf(acc.w, v.w);
    }

    // Streaming store: write-once output must not evict the L2-resident table.
    v4f* __restrict__ orow = (v4f*)(out + (size_t)node * EMBED_DIM);
    __builtin_nontemporal_store(acc, orow + lane);
}

extern "C" void kernel_launch(void* const* d_in, const int* in_sizes, int n_in,
                              void* d_out, int out_size, void* d_ws, size_t ws_size,
                              hipStream_t stream) {
    (void)in_sizes; (void)n_in; (void)d_ws; (void)ws_size;

    const float* table = (const float*)d_in[0];   // [200000, 128] fp32
    const int*   ids   = (const int*)d_in[1];     // [1.6M] hyperedge ids
    // d_in[2] (segment_ids) is implied by layout: entry i belongs to node i/16.
    // d_in[3] (n_nodes scalar) lives on device; derive it from out_size instead
    // (graph capture forbids a sync copy-back).
    float* out = (float*)d_out;

    const int n_nodes = out_size / EMBED_DIM;
    const int grid = (n_nodes + WAVES_PER_BLOCK - 1) / WAVES_PER_BLOCK;

    node_max_aggregate_kernel<<<grid, BLOCK_THREADS, 0, stream>>>(
        table, ids, out, n_nodes);
}